// RelativeSAMAttention_40973988004772
// MI455X (gfx1250) — compile-verified
//
#include <hip/hip_runtime.h>

typedef __attribute__((ext_vector_type(2))) float v2f;
typedef __attribute__((ext_vector_type(4))) float v4f;
typedef __attribute__((ext_vector_type(8))) float v8f;

#define B_   4
#define H_   8
#define N_   1024
#define DH_  128
#define ROWS 32

__device__ __forceinline__ float softplusf(float x) {
    float ax = __builtin_fabsf(x);
    return fmaxf(x, 0.0f) + __logf(1.0f + __expf(-ax));
}

extern "C" __global__ __launch_bounds__(256)
void rel_sam_attn(const float* __restrict__ q,  const float* __restrict__ k,
                  const float* __restrict__ v,  const float* __restrict__ c,
                  const float* __restrict__ dq, const float* __restrict__ dk,
                  const float* __restrict__ w_w, const float* __restrict__ b_w,
                  const float* __restrict__ w_b, const float* __restrict__ b_b,
                  float* __restrict__ out, float* __restrict__ scores)
{
    extern __shared__ float smem[];
    float* s_buf   = smem;              // ROWS * N_ floats (128 KB score tile)
    float* rowsum  = smem + ROWS * N_;  // 32
    float* inv_det = rowsum + ROWS;     // 32
    float* dq3s    = inv_det + ROWS;    // 32*4 (x,y,z,|.|^2 per row)
    float* csum    = dq3s + ROWS * 4;   // 1

    const int tid  = threadIdx.x;
    const int b    = blockIdx.z;
    const int h    = blockIdx.y;
    const int row0 = blockIdx.x * ROWS;
    const size_t bh = (size_t)b * H_ + h;

    // ---------------- phase 0: init + per-row dq3 data + sum softplus(c) ----
    if (tid < ROWS) {
        rowsum[tid] = 0.0f;
        const float* dqp = dq + ((size_t)b * N_ + row0 + tid) * 4;
        float x = dqp[0], y = dqp[1], z = dqp[2];
        dq3s[tid*4+0] = x; dq3s[tid*4+1] = y; dq3s[tid*4+2] = z;
        dq3s[tid*4+3] = x*x + y*y + z*z;
    }
    if (tid == 0) csum[0] = 0.0f;
    __syncthreads();
    {
        const float* cp = c + bh * N_;
        float lc = 0.0f;
        for (int m = tid; m < N_; m += 256) lc += softplusf(cp[m]);
        atomicAdd(csum, lc);
    }

    const float ww = w_w[h], bw = b_w[h], wb = w_b[h], bb = b_b[h];
    const float scale = 0.088388347648318447f; // 1/sqrt(128)

    const int wv     = tid >> 5;
    const int lane   = tid & 31;
    const int lhalf  = lane >> 4;   // 0/1 half-wave
    const int l16    = lane & 15;
    const int rsub   = wv & 1;      // which 16-row subtile
    const int mslice = wv >> 1;     // m-tile slice (4 waves per row subtile)

    // ---------------- phase 1: S = relu((QK^T/sqrt(dh))*sp(-(d*ww+bw)) + d*wb+bb)
    // A fragments (q rows) kept resident across the whole m sweep.
    v2f afrag[32];
    {
        const float* qrow = q + (bh * N_ + row0 + rsub*16 + l16) * DH_;
        #pragma unroll
        for (int kk = 0; kk < 32; ++kk)
            afrag[kk] = *(const v2f*)(qrow + kk*4 + 2*lhalf);
    }
    // row geometry for the 8 rows this lane's accumulator covers
    float rx[8], ry[8], rz[8], rn2[8], rs[8];
    #pragma unroll
    for (int j = 0; j < 8; ++j) {
        int r = rsub*16 + j + 8*lhalf;
        rx[j]  = dq3s[r*4+0]; ry[j] = dq3s[r*4+1];
        rz[j]  = dq3s[r*4+2]; rn2[j] = dq3s[r*4+3];
        rs[j]  = 0.0f;
    }

    for (int mt = mslice; mt < N_/16; mt += 4) {
        const int mbase = mt * 16;
        const float* krow = k + (bh * N_ + mbase + l16) * DH_;
        v8f acc = {};
        #pragma unroll
        for (int kk = 0; kk < 32; ++kk) {
            v2f bfrag = *(const v2f*)(krow + kk*4 + 2*lhalf);
            acc = __builtin_amdgcn_wmma_f32_16x16x4_f32(false, afrag[kk], false, bfrag,
                                                        (short)0, acc, false, false);
        }
        // fused epilogue: dist map + relu; write tile to LDS, keep row sums
        const float* dkp = dk + ((size_t)b * N_ + mbase + l16) * 4;
        float kx = dkp[0], ky = dkp[1], kz = dkp[2];
        float kn2 = kx*kx + ky*ky + kz*kz;
        #pragma unroll
        for (int j = 0; j < 8; ++j) {
            int r = rsub*16 + j + 8*lhalf;
            float dist = rn2[j] + kn2 - 2.0f*(rx[j]*kx + ry[j]*ky + rz[j]*kz);
            float dw   = softplusf(-(dist*ww + bw));
            float dbv  = dist*wb + bb;
            float sv   = fmaxf(acc[j]*scale*dw + dbv, 0.0f);
            rs[j] += sv;
            s_buf[r*N_ + mbase + l16] = sv;
        }
    }
    #pragma unroll
    for (int j = 0; j < 8; ++j)
        atomicAdd(&rowsum[rsub*16 + j + 8*lhalf], rs[j]);
    __syncthreads();

    // ---------------- phase 2: denominator ----------------
    if (tid < ROWS) inv_det[tid] = 1.0f / (rowsum[tid] + csum[0] + 1e-9f);
    __syncthreads();

    // ---------------- phase 3: single normalized NT write of scores --------
    // scores is a 134MB one-shot stream; mark non-temporal so it does not
    // evict the heavily reused K/V/Q working set (~48MB) from the 192MB L2.
    {
        float* srow = scores + (bh * N_ + row0) * (size_t)N_;
        for (int i = tid; i < ROWS*N_/4; i += 256) {
            int idx = i * 4;
            int r = idx >> 10;
            float inv = inv_det[r];
            v4f o;
            o[0] = s_buf[idx+0] * inv;
            o[1] = s_buf[idx+1] * inv;
            o[2] = s_buf[idx+2] * inv;
            o[3] = s_buf[idx+3] * inv;
            __builtin_nontemporal_store(o, (v4f*)(srow + idx)); // row stride == N_
        }
    }

    // ---------------- phase 4: out = S_norm @ V (A from LDS, K = 1024) ----
    #pragma unroll
    for (int t = 0; t < 2; ++t) {
        const int rsub2 = t;         // 16-row subtile
        const int dt    = wv;        // 16-wide d column tile (8 waves -> DH=128)
        const float* vb    = v + bh * N_ * DH_ + dt*16 + l16;
        const float* abase = s_buf + (size_t)(rsub2*16 + l16) * N_;
        v8f acc = {};
        for (int kk = 0; kk < N_/4; ++kk) {
            int k0 = kk*4 + 2*lhalf;
            v2f af; af[0] = abase[k0];             af[1] = abase[k0+1];
            v2f bf; bf[0] = vb[(size_t)k0 * DH_];  bf[1] = vb[(size_t)(k0+1) * DH_];
            acc = __builtin_amdgcn_wmma_f32_16x16x4_f32(false, af, false, bf,
                                                        (short)0, acc, false, false);
        }
        #pragma unroll
        for (int j = 0; j < 8; ++j) {
            int r = rsub2*16 + j + 8*lhalf;
            float o = acc[j] * inv_det[r];
            __builtin_nontemporal_store(
                o, out + (bh * N_ + row0 + r) * DH_ + dt*16 + l16);
        }
    }
}

extern "C" void kernel_launch(void* const* d_in, const int* in_sizes, int n_in,
                              void* d_out, int out_size, void* d_ws, size_t ws_size,
                              hipStream_t stream) {
    const float* q   = (const float*)d_in[0];
    const float* k   = (const float*)d_in[1];
    const float* v   = (const float*)d_in[2];
    const float* c   = (const float*)d_in[3];
    const float* dq  = (const float*)d_in[4];
    const float* dk  = (const float*)d_in[5];
    const float* w_w = (const float*)d_in[6];
    const float* b_w = (const float*)d_in[7];
    const float* w_b = (const float*)d_in[8];
    const float* b_b = (const float*)d_in[9];

    float* out    = (float*)d_out;
    float* scores = out + (size_t)B_ * H_ * N_ * DH_;

    const size_t smem = (size_t)(ROWS * N_ + ROWS + ROWS + ROWS*4 + 1) * sizeof(float);
    // Opt in to >64KB dynamic LDS (CDNA5 WGP has 320KB); ignore result if no-op.
    (void)hipFuncSetAttribute((const void*)rel_sam_attn,
                              hipFuncAttributeMaxDynamicSharedMemorySize, (int)smem);

    dim3 grid(N_ / ROWS, H_, B_), block(256);
    hipLaunchKernelGGL(rel_sam_attn, grid, block, smem, stream,
                       q, k, v, c, dq, dk, w_w, b_w, w_b, b_b, out, scores);
}